// TdoaLoss_5214090297528
// MI455X (gfx1250) — compile-verified
//
#include <hip/hip_runtime.h>
#include <stdint.h>

#define BDIM    64
#define CELLS   32              // (b,t) cells per block
#define PRED_F  234             // 13*3*6 floats per cell
#define TARG_F  195             // 3*5*13 floats per cell
#define PRED_LDS_F (CELLS * PRED_F)   // 7488 floats = 29952 B
#define TARG_LDS_F (CELLS * TARG_F)   // 6240 floats = 24960 B
#define IGN     (-100)
#define FBDIM   256

__device__ __forceinline__ float sel3f(float a, float b, float c, int r) {
  return r == 0 ? a : (r == 1 ? b : c);
}
__device__ __forceinline__ int sel3i(int a, int b, int c, int r) {
  return r == 0 ? a : (r == 1 ? b : c);
}

__global__ void tdoa_init(unsigned int* counters) {
  if (threadIdx.x < 4) counters[threadIdx.x] = 0u;
}

__global__ __launch_bounds__(BDIM) void tdoa_main(
    const float* __restrict__ pred, const float* __restrict__ targ,
    float* __restrict__ partial, unsigned int* __restrict__ counters) {
  __shared__ __align__(16) float smem[PRED_LDS_F + TARG_LDS_F];
  __shared__ float r_loss[BDIM];
  __shared__ unsigned r_nv[BDIM], r_cc[BDIM], r_np[BDIM];

  const int tid = threadIdx.x;
  const size_t cell0 = (size_t)blockIdx.x * CELLS;

  // ---- Stage this block's tile into LDS with CDNA5 async global->LDS DMA.
  //      Single-pass streamed data -> non-temporal hint (don't pollute GL2).
  {
    const float* gp = pred + cell0 * PRED_F;   // contiguous 29952 B
    const float* gt = targ + cell0 * TARG_F;   // contiguous 24960 B
    unsigned ldsPred = (unsigned)(uintptr_t)(&smem[0]);
    unsigned ldsTarg = (unsigned)(uintptr_t)(&smem[PRED_LDS_F]);
    const int PCH = PRED_LDS_F / 4;  // 1872 x 16B chunks
    const int TCH = TARG_LDS_F / 4;  // 1560 x 16B chunks
    for (int i = tid; i < PCH; i += BDIM) {
      unsigned off = (unsigned)i * 16u;
      unsigned lds = ldsPred + off;
      asm volatile("global_load_async_to_lds_b128 %0, %1, %2 th:TH_LOAD_NT"
                   :: "v"(lds), "v"(off), "s"(gp) : "memory");
    }
    for (int i = tid; i < TCH; i += BDIM) {
      unsigned off = (unsigned)i * 16u;
      unsigned lds = ldsTarg + off;
      asm volatile("global_load_async_to_lds_b128 %0, %1, %2 th:TH_LOAD_NT"
                   :: "v"(lds), "v"(off), "s"(gt) : "memory");
    }
    asm volatile("s_wait_asynccnt 0" ::: "memory");
  }
  __syncthreads();

  const int lcell = tid >> 1;      // 0..31
  const int half  = tid & 1;       // ntd slots [3*half, 3*half+3)
  const float* tp = &smem[PRED_LDS_F + lcell * TARG_F];
  const float* pp = &smem[lcell * PRED_F];

  // ---- Per-cell TDOA targets (only this thread's 3 pair slots) ----
  const float MICX[4] = { 0.024327574f,  0.024327574f, -0.024327574f, -0.024327574f};
  const float MICY[4] = { 0.024327574f, -0.024327574f,  0.024327574f, -0.024327574f};
  const float MICZ[4] = { 0.024090210f, -0.024090210f, -0.024090210f,  0.024090210f};

  float td[3][3];   // [track e][local pair q], static indexing only
  bool act[3];
  #pragma unroll
  for (int e = 0; e < 3; ++e) {
    const float* tb_ = tp + e * 65;     // e stride = 5*13
    float d0 = 0.f, d1 = 0.f, d2 = 0.f, ds = 0.f;
    bool a = false;
    #pragma unroll
    for (int c = 0; c < 13; ++c) {
      float av = tb_[c];
      a = a || (av > 0.0f);
      d0 += av * tb_[13 + c];
      d1 += av * tb_[26 + c];
      d2 += av * tb_[39 + c];
      ds += av * tb_[52 + c];
    }
    act[e] = a;
    float sx = d0 * ds, sy = d1 * ds, sz = d2 * ds;
    float dm[4];
    #pragma unroll
    for (int m = 0; m < 4; ++m) {
      float dx = sx - MICX[m], dy = sy - MICY[m], dz = sz - MICZ[m];
      dm[m] = sqrtf(dx * dx + dy * dy + dz * dz);
    }
    // PAIRS: half=0 -> (0,1)(0,2)(0,3); half=1 -> (1,2)(1,3)(2,3)
    #pragma unroll
    for (int q = 0; q < 3; ++q) {
      float dA = half ? (q == 2 ? dm[2] : dm[1]) : dm[0];
      float dB = half ? (q == 0 ? dm[2] : dm[3]) : dm[q + 1];
      td[e][q] = rintf((dA - dB) * 24000.0f / 343.0f) + 6.0f;
    }
  }

  const int nact = (int)act[0] + (int)act[1] + (int)act[2];
  // stable sort by (!active): closed-form order (only used when nact>=1)
  const int ord0 = act[0] ? 0 : (act[1] ? 1 : 2);
  const int ord1 = act[0] ? (act[1] ? 1 : 2) : 2;
  const int ord2 = 2;
  const int ra0 = ord0;
  const int ra1 = (nact >= 2) ? ord1 : ord0;
  const int ra2 = (nact == 3) ? ord2 : ((nact == 2) ? ord1 : ord0);
  const int rb2 = (nact == 3) ? ord2 : ord0;

  float lsum = 0.0f;
  unsigned nv = 0, cc = 0, np = 0;

  #pragma unroll
  for (int q = 0; q < 3; ++q) {
    const int ntd = half * 3 + q;

    // targets t_a / t_b for this ntd
    int tA0, tA1, tA2, tB0, tB1, tB2;
    if (nact == 0) {
      tA0 = tA1 = tA2 = tB0 = tB1 = tB2 = IGN;
    } else {
      float f0 = td[0][q], f1 = td[1][q], f2 = td[2][q];
      tA0 = (int)sel3f(f0, f1, f2, ra0);
      tA1 = (int)sel3f(f0, f1, f2, ra1);
      tA2 = (int)sel3f(f0, f1, f2, ra2);
      tB0 = tA0;                         // idx2(0)==idx1(0)
      tB1 = tA1;                         // idx2(1)==idx1(1)
      tB2 = (int)sel3f(f0, f1, f2, rb2);
    }

    // logits P[c][j] = pred[b,t,c,j,ntd]
    float P[13][3];
    #pragma unroll
    for (int c = 0; c < 13; ++c)
      #pragma unroll
      for (int j = 0; j < 3; ++j)
        P[c][j] = pp[c * 18 + j * 6 + ntd];

    float mxv[3], lg[3];
    int am[3];
    #pragma unroll
    for (int j = 0; j < 3; ++j) {
      float m = P[0][j]; int mi = 0;
      #pragma unroll
      for (int c = 1; c < 13; ++c)
        if (P[c][j] > m) { m = P[c][j]; mi = c; }   // first max, like jnp.argmax
      float s = 0.0f;
      #pragma unroll
      for (int c = 0; c < 13; ++c) s += __expf(P[c][j] - m);   // v_exp_f32
      mxv[j] = m; lg[j] = __logf(s); am[j] = mi;               // v_log_f32
    }

    // L[i][j] = -logp[tc[i]][j], masked; class gather straight from LDS
    const int tAv[3] = {tA0, tA1, tA2};
    const int tBv[3] = {tB0, tB1, tB2};
    float L1[3][3], L2[3][3];
    #pragma unroll
    for (int i = 0; i < 3; ++i) {
      int t1 = tAv[i], t2 = tBv[i];
      int c1 = t1 < 0 ? 0 : (t1 > 12 ? 12 : t1);
      int c2 = t2 < 0 ? 0 : (t2 > 12 ? 12 : t2);
      #pragma unroll
      for (int j = 0; j < 3; ++j) {
        float p1 = pp[c1 * 18 + j * 6 + ntd];
        float p2 = pp[c2 * 18 + j * 6 + ntd];
        L1[i][j] = (t1 == IGN) ? 0.0f : (lg[j] - (p1 - mxv[j]));
        L2[i][j] = (t2 == IGN) ? 0.0f : (lg[j] - (p2 - mxv[j]));
      }
    }

    // 6-permutation PIT, first-min argmin
    const int PR[6][3] = {{0,1,2},{0,2,1},{1,0,2},{1,2,0},{2,0,1},{2,1,0}};
    float b1 = (L1[0][0] + L1[1][1] + L1[2][2]) / 3.0f;
    float b2 = (L2[0][0] + L2[1][1] + L2[2][2]) / 3.0f;
    int bp1 = 0, bp2 = 0;
    #pragma unroll
    for (int p = 1; p < 6; ++p) {
      float v1 = (L1[0][PR[p][0]] + L1[1][PR[p][1]] + L1[2][PR[p][2]]) / 3.0f;
      float v2 = (L2[0][PR[p][0]] + L2[1][PR[p][1]] + L2[2][PR[p][2]]) / 3.0f;
      if (v1 < b1) { b1 = v1; bp1 = p; }
      if (v2 < b2) { b2 = v2; bp2 = p; }
    }

    const bool use2 = (b2 < b1);
    const float loss = use2 ? b2 : b1;
    const int bp = use2 ? bp2 : bp1;
    const int ts0 = use2 ? tB0 : tA0;
    const int ts1 = use2 ? tB1 : tA1;
    const int ts2 = use2 ? tB2 : tA2;

    // decode PERM[bp] without table lookup
    const int x   = bp >> 1;
    const int odd = bp & 1;
    const int sm  = (x == 0) ? 1 : 0;
    const int lgr = (x == 2) ? 1 : 2;
    const int pj0 = x;
    const int pj1 = odd ? lgr : sm;
    const int pj2 = odd ? sm : lgr;

    const int t0 = sel3i(ts0, ts1, ts2, pj0);
    const int t1 = sel3i(ts0, ts1, ts2, pj1);
    const int t2 = sel3i(ts0, ts1, ts2, pj2);
    if (t0 != IGN) { ++np; if (am[0] == t0) ++cc; }
    if (t1 != IGN) { ++np; if (am[1] == t1) ++cc; }
    if (t2 != IGN) { ++np; if (am[2] == t2) ++cc; }
    if (loss > 0.0f) { lsum += loss; ++nv; }
  }

  // ---- block reduction (fixed order -> deterministic) ----
  r_loss[tid] = lsum; r_nv[tid] = nv; r_cc[tid] = cc; r_np[tid] = np;
  __syncthreads();
  if (tid == 0) {
    float s = 0.0f; unsigned a = 0, b = 0, c = 0;
    for (int i = 0; i < BDIM; ++i) { s += r_loss[i]; a += r_nv[i]; b += r_cc[i]; c += r_np[i]; }
    partial[blockIdx.x] = s;
    atomicAdd(&counters[0], a);
    atomicAdd(&counters[1], b);
    atomicAdd(&counters[2], c);
  }
}

__global__ __launch_bounds__(FBDIM) void tdoa_final(
    const float* __restrict__ partial, int nblocks,
    const unsigned int* __restrict__ counters, float* __restrict__ out) {
  __shared__ float sh[FBDIM];
  float s = 0.0f;
  for (int i = threadIdx.x; i < nblocks; i += FBDIM) s += partial[i];
  sh[threadIdx.x] = s;
  __syncthreads();
  if (threadIdx.x == 0) {
    float tot = 0.0f;
    for (int i = 0; i < FBDIM; ++i) tot += sh[i];
    unsigned nval = counters[0], correct = counters[1], npred = counters[2];
    out[0] = (nval > 0) ? tot / (float)nval : 0.0f;
    out[1] = (npred > 0) ? (float)correct / (float)npred : 0.0f;
  }
}

extern "C" void kernel_launch(void* const* d_in, const int* in_sizes, int n_in,
                              void* d_out, int out_size, void* d_ws, size_t ws_size,
                              hipStream_t stream) {
  (void)n_in; (void)out_size; (void)ws_size;
  const float* pred = (const float*)d_in[0];
  const float* targ = (const float*)d_in[1];
  float* out = (float*)d_out;

  const int cells   = in_sizes[0] / PRED_F;   // B*T = 76800
  const int nblocks = cells / CELLS;          // 2400

  unsigned char* ws = (unsigned char*)d_ws;
  unsigned int* counters = (unsigned int*)ws;        // 16 B
  float* partial = (float*)(ws + 16);                // nblocks floats

  tdoa_init<<<1, 32, 0, stream>>>(counters);
  tdoa_main<<<nblocks, BDIM, 0, stream>>>(pred, targ, partial, counters);
  tdoa_final<<<1, FBDIM, 0, stream>>>(partial, nblocks, counters, out);
}